// SelectiveScan_65755949302282
// MI455X (gfx1250) — compile-verified
//
#include <hip/hip_runtime.h>
#include <hip/hip_bf16.h>

#define BATCH   2
#define LSEQ    1024
#define DMODEL  1024
#define DSTATE  16
#define DCONV   4
#define DINNER  2048

typedef float v2f __attribute__((ext_vector_type(2)));
typedef float v8f __attribute__((ext_vector_type(8)));

// ---------------------------------------------------------------------------
// Generic fp32 WMMA GEMM:  C[M,N] = A[M,K] x B[K,N]  (+ optional epilogue)
//   MT       : number of 16-row output tiles per wave (compile-time)
//   NT       : number of 16-col output tiles per wave (compile-time)
//   BT=false : B row-major, element (k,n) at Bm[k*N + n]
//   BT=true  : Bm is W[N,K] row-major, element (k,n) at Bm[n*K + k]  (B = W^T)
//   EPI=0    : plain store
//   EPI=1    : val = softplus(val + bias[n])
// Block = 128 threads (4 waves). Wave w covers rows [m0, m0+16*MT), cols 16*NT.
// Each B operand load feeds MT WMMAs (B k-slices are shared across M-tiles),
// so MT=2 halves B traffic per output vs MT=1.
// Uses V_WMMA_F32_16X16X4_F32 (wave32): A 16x4, B 4x16, C/D 16x16 fp32.
//   A lane layout: lane&15 = M, (lane>>4) selects K-pair {0,1}/{2,3}
//   B lane layout: lane&15 = N, (lane>>4) selects K-pair
//   C/D: lane&15 = N, VGPR v -> M = v + 8*(lane>>4)
// Straight-line K-loop body: all MT+NT loads issued before the MT*NT WMMAs;
// x2 unroll lets next-iteration loads overlap current WMMAs.
// ---------------------------------------------------------------------------
template<int MT, int NT, int EPI, bool BT>
__global__ __launch_bounds__(128)
void gemm16x4_wmma(const float* __restrict__ A, const float* __restrict__ Bm,
                   const float* __restrict__ bias, float* __restrict__ C,
                   int M, int N, int K) {
    const int lane = threadIdx.x & 31;
    const int wv   = threadIdx.x >> 5;
    const int m0   = blockIdx.y * (64 * MT) + wv * (16 * MT);
    const int n0   = blockIdx.x * (16 * NT);
    const int ml   = lane & 15;
    const int hi   = lane >> 4;          // 0 or 1 (wave-half)

    v8f acc[MT][NT] = {};

    const float* ap = A + (size_t)(m0 + ml) * K + (hi << 1);

#pragma unroll 2
    for (int k = 0; k < K; k += 4) {
        // ---- issue all loads for this K-step first ----
        v2f a[MT];
#pragma unroll
        for (int i = 0; i < MT; ++i) {
            a[i] = *(const v2f*)(ap + (size_t)(16 * i) * K + k);
        }
        v2f b[NT];
#pragma unroll
        for (int t = 0; t < NT; ++t) {
            const int n = n0 + t * 16 + ml;
            if (BT) {
                b[t] = *(const v2f*)(Bm + (size_t)n * K + k + (hi << 1));
            } else {
                const size_t kb = (size_t)(k + (hi << 1));
                b[t].x = Bm[kb * N + n];
                b[t].y = Bm[(kb + 1) * N + n];
            }
        }
        // ---- then the MT*NT matrix ops (each b[t] reused MT times) ----
#pragma unroll
        for (int t = 0; t < NT; ++t) {
#pragma unroll
            for (int i = 0; i < MT; ++i) {
                acc[i][t] = __builtin_amdgcn_wmma_f32_16x16x4_f32(
                    /*neg_a=*/false, a[i], /*neg_b=*/false, b[t],
                    /*c_mod=*/(short)0, acc[i][t],
                    /*reuse_a=*/false, /*reuse_b=*/false);
            }
        }
    }

#pragma unroll
    for (int i = 0; i < MT; ++i) {
#pragma unroll
        for (int t = 0; t < NT; ++t) {
            const int n = n0 + t * 16 + ml;
#pragma unroll
            for (int v = 0; v < 8; ++v) {
                const int row = m0 + 16 * i + v + (hi << 3);
                float val = acc[i][t][v];
                if (EPI == 1) {
                    // numerically stable softplus(val + bias[n])
                    const float xv = val + bias[n];
                    val = fmaxf(xv, 0.f) + log1pf(__expf(-fabsf(xv)));
                }
                C[(size_t)row * N + n] = val;
            }
        }
    }
}

// ---------------------------------------------------------------------------
// Depthwise causal conv (K=4) + bias + SiLU over the first DINNER lanes of xr.
// xr: (B*L, 2*DINNER) row-major; u out: (B*L, DINNER)
// ---------------------------------------------------------------------------
__global__ __launch_bounds__(256)
void conv_silu_kernel(const float* __restrict__ xr, const float* __restrict__ cw,
                      const float* __restrict__ cb, float* __restrict__ u) {
    const int id = blockIdx.x * blockDim.x + threadIdx.x;   // over B*L*DINNER
    const int d  = id & (DINNER - 1);
    const int r  = id >> 11;            // global row in [0, B*L)
    const int b  = r / LSEQ;
    const int l  = r - b * LSEQ;

    float acc = cb[d];
#pragma unroll
    for (int t = 0; t < DCONV; ++t) {
        const int ls = l - (DCONV - 1) + t;
        if (ls >= 0) {
            acc += xr[((size_t)(b * LSEQ + ls)) * (2 * DINNER) + d] * cw[d * DCONV + t];
        }
    }
    // SiLU
    const float s = acc / (1.f + __expf(-acc));
    u[(size_t)r * DINNER + d] = s;
}

// ---------------------------------------------------------------------------
// Selective scan. One lane per (channel, state): 16 lanes cooperate on one
// (b,d) channel; the 16-wide cross-lane sum uses ds-permute shuffles.
// Fuses: h = exp(dt*A)*h + dt*B*u ; y = sum_s h*C ; y += u*D ; y *= silu(res)
// Writes result IN PLACE over the delta buffer (read-before-write per step).
// ---------------------------------------------------------------------------
__global__ __launch_bounds__(256)
void scan_kernel(const float* __restrict__ delta, const float* __restrict__ u,
                 const float* __restrict__ xdbl, const float* __restrict__ xr,
                 const float* __restrict__ A_log, const float* __restrict__ Dp,
                 float* __restrict__ yg) {
    const int tid = blockIdx.x * blockDim.x + threadIdx.x;  // 65536 lanes
    const int s   = tid & (DSTATE - 1);
    const int ch  = tid >> 4;                               // [0, B*DINNER)
    const int b   = ch >> 11;                               // ch / DINNER
    const int d   = ch & (DINNER - 1);

    const float As = -__expf(A_log[d * DSTATE + s]);
    const float Dv = Dp[d];

    float h = 0.f;
    const size_t rbase = (size_t)b * LSEQ;
    for (int l = 0; l < LSEQ; ++l) {
        const size_t row = rbase + l;
        const float dt = delta[row * DINNER + d];   // broadcast across 16 lanes
        const float uu = u[row * DINNER + d];       // broadcast
        const float Bv = xdbl[row * (2 * DSTATE) + s];
        const float Cv = xdbl[row * (2 * DSTATE) + DSTATE + s];

        h = __expf(dt * As) * h + dt * Bv * uu;
        float t = h * Cv;
        // 16-wide xor-tree reduction (stays inside each half-wave)
        t += __shfl_xor(t, 1, 16);
        t += __shfl_xor(t, 2, 16);
        t += __shfl_xor(t, 4, 16);
        t += __shfl_xor(t, 8, 16);

        if (s == 0) {
            const float res = xr[row * (2 * DINNER) + DINNER + d];
            const float y   = t + uu * Dv;
            yg[row * DINNER + d] = y * (res / (1.f + __expf(-res)));
        }
    }
}

// ---------------------------------------------------------------------------
extern "C" void kernel_launch(void* const* d_in, const int* in_sizes, int n_in,
                              void* d_out, int out_size, void* d_ws, size_t ws_size,
                              hipStream_t stream) {
    const float* x       = (const float*)d_in[0];  // (B, L, DMODEL)
    const float* in_w    = (const float*)d_in[1];  // (DMODEL, 2*DINNER)
    const float* conv_w  = (const float*)d_in[2];  // (DINNER, 1, DCONV)
    const float* conv_b  = (const float*)d_in[3];  // (DINNER,)
    const float* xproj_w = (const float*)d_in[4];  // (DINNER, 2*DSTATE)
    const float* dt_w    = (const float*)d_in[5];  // (DINNER, DINNER)
    const float* dt_b    = (const float*)d_in[6];  // (DINNER,)
    const float* A_log   = (const float*)d_in[7];  // (DINNER, DSTATE)
    const float* Dp      = (const float*)d_in[8];  // (DINNER,)
    const float* out_w   = (const float*)d_in[9];  // (DINNER, DMODEL)
    float* out = (float*)d_out;                    // (B, L, DMODEL)

    const int BL = BATCH * LSEQ;                   // 2048 rows

    // Workspace layout (bytes):
    char* ws = (char*)d_ws;
    float* xr    = (float*)(ws);                         // 2048*4096*4 = 32 MB
    float* u     = (float*)(ws + (size_t)33554432);      // 2048*2048*4 = 16 MB
    float* xdbl  = (float*)(ws + (size_t)50331648);      // 2048*32*4   = 256 KB
    float* delta = (float*)(ws + (size_t)50593792);      // 16 MB (reused as yg)

    dim3 blk(128);

    // 1) xr = x @ in_proj_w        M=2048, K=1024, N=4096
    gemm16x4_wmma<2, 4, 0, false><<<dim3(4096 / 64, BL / 128), blk, 0, stream>>>(
        x, in_w, nullptr, xr, BL, 2 * DINNER, DMODEL);

    // 2) u = silu(causal_conv(xr[:, :DINNER]) + conv_b)
    conv_silu_kernel<<<(BL * DINNER) / 256, 256, 0, stream>>>(xr, conv_w, conv_b, u);

    // 3) x_dbl = u @ x_proj_w      M=2048, K=2048, N=32  (NT=2 covers 32 cols)
    gemm16x4_wmma<2, 2, 0, false><<<dim3(1, BL / 128), blk, 0, stream>>>(
        u, xproj_w, nullptr, xdbl, BL, 2 * DSTATE, DINNER);

    // 4) delta = softplus(u @ dt_proj_w^T + dt_proj_b)   M=2048, K=2048, N=2048
    gemm16x4_wmma<2, 4, 1, true><<<dim3(DINNER / 64, BL / 128), blk, 0, stream>>>(
        u, dt_w, dt_b, delta, BL, DINNER, DINNER);

    // 5) selective scan (+ u*D + silu(res) gating), in place over delta
    scan_kernel<<<(BATCH * DINNER * DSTATE) / 256, 256, 0, stream>>>(
        delta, u, xdbl, xr, A_log, Dp, delta);

    // 6) out = yg @ out_proj_w     M=2048, K=2048, N=1024
    gemm16x4_wmma<2, 4, 0, false><<<dim3(DMODEL / 64, BL / 128), blk, 0, stream>>>(
        delta, out_w, nullptr, out, BL, DMODEL, DINNER);
}